// OTResampler_41455024341461
// MI455X (gfx1250) — compile-verified
//
#include <hip/hip_runtime.h>

// Problem constants (match reference)
#define BB 16
#define NN 1024
#define DDIM 4
#define EPS_F 0.1f
#define SCALING_F 0.9f
#define THRESH_F 0.001f
#define MAX_ITER_I 100

#define NT 256      // threads per block (8 wave32s)
#define NWAVE 8
#define NIB 64      // 1024 / 16 tiles
#define FBIG 3.402823466e38f

typedef __attribute__((ext_vector_type(2))) float v2f;
typedef __attribute__((ext_vector_type(8))) float v8f;

// Gram tile: D += A(16x4) * B(4x16), f32, via CDNA5 WMMA (K=4 == particle dim)
__device__ __forceinline__ v8f wmma_gram(v2f a, v2f b, v8f c) {
  return __builtin_amdgcn_wmma_f32_16x16x4_f32(false, a, false, b, (short)0, c,
                                               false, false);
}

__device__ __forceinline__ float block_sum(float v, float* scr, int tid) {
#pragma unroll
  for (int off = 16; off; off >>= 1) v += __shfl_xor(v, off, 32);
  if ((tid & 31) == 0) scr[tid >> 5] = v;
  __syncthreads();
  if (tid == 0) {
    float s = 0.f;
#pragma unroll
    for (int w = 0; w < NWAVE; ++w) s += scr[w];
    scr[0] = s;
  }
  __syncthreads();
  float r = scr[0];
  __syncthreads();
  return r;
}

__device__ __forceinline__ float block_max(float v, float* scr, int tid) {
#pragma unroll
  for (int off = 16; off; off >>= 1) v = fmaxf(v, __shfl_xor(v, off, 32));
  if ((tid & 31) == 0) scr[tid >> 5] = v;
  __syncthreads();
  if (tid == 0) {
    float s = -FBIG;
#pragma unroll
    for (int w = 0; w < NWAVE; ++w) s = fmaxf(s, scr[w]);
    scr[0] = s;
  }
  __syncthreads();
  float r = scr[0];
  __syncthreads();
  return r;
}

// out[i] = -eps * log sum_j exp(f[j] - C[i,j]/eps),
// C[i,j] = 0.5*(n[i]+n[j]) - <sx_i, sx_j>, Gram tile via WMMA.
// fM = max_j f[j] is a valid stabilizer because C >= 0.
// Caller must __syncthreads() after (out written by lane lh==0 only).
__device__ void softmin_lds(const float* __restrict__ sxl,
                            const float* __restrict__ nl,
                            const float* __restrict__ f, float fM,
                            float* __restrict__ outp, float eps, int tid) {
  const int lane = tid & 31;
  const int wave = tid >> 5;
  const int lh = lane & 15;   // lane within half
  const int hi = lane >> 4;   // which half
  const int ko = hi * 2;      // K offset carried by this half (A/B frag layout)
  const float inv_eps = 1.0f / eps;

  for (int ib = wave; ib < NIB; ib += NWAVE) {
    const int i0 = ib * 16;
    const int arow = i0 + lh;
    v2f a;
    a.x = sxl[arow * 4 + ko];
    a.y = sxl[arow * 4 + ko + 1];
    const int rbase = i0 + hi * 8;  // rows owned by slots v=0..7 in this half
    float nrow[8];
#pragma unroll
    for (int v = 0; v < 8; ++v) nrow[v] = nl[rbase + v];
    float s[8];
#pragma unroll
    for (int v = 0; v < 8; ++v) s[v] = 0.0f;

    for (int jb = 0; jb < NIB; ++jb) {
      const int col = jb * 16 + lh;
      v2f bf;
      bf.x = sxl[col * 4 + ko];
      bf.y = sxl[col * 4 + ko + 1];
      v8f g = {0.f, 0.f, 0.f, 0.f, 0.f, 0.f, 0.f, 0.f};
      g = wmma_gram(a, bf, g);
      const float ncol = nl[col];
      const float fc = f[col] - fM;
#pragma unroll
      for (int v = 0; v < 8; ++v) {
        const float cost = 0.5f * (nrow[v] + ncol) - g[v];
        s[v] += __expf(fc - cost * inv_eps);
      }
    }
    // reduce across the 16 lanes of each half (columns of the tile row)
#pragma unroll
    for (int off = 8; off; off >>= 1) {
#pragma unroll
      for (int v = 0; v < 8; ++v) s[v] += __shfl_xor(s[v], off, 32);
    }
    if (lh == 0) {
#pragma unroll
      for (int v = 0; v < 8; ++v)
        outp[rbase + v] = -eps * (fM + __logf(s[v]));
    }
  }
}

__global__ __launch_bounds__(NT) void ot_sinkhorn_kernel(
    const float* __restrict__ particles, const float* __restrict__ weights,
    const float* __restrict__ logw_g, const unsigned char* __restrict__ flags,
    float* __restrict__ out) {
  __shared__ float sxl[NN * 4];  // normalized particles, row-major [n][d]
  __shared__ float nl[NN];       // squared norms of sx rows
  __shared__ float la[NN];       // log weights
  __shared__ float ay[NN], bx[NN], ax[NN], by[NN];
  __shared__ float t1[NN], t2[NN];
  __shared__ float fbuf[NN];
  __shared__ float scr[NWAVE];

  const int b = blockIdx.x;
  const int tid = threadIdx.x;
  const int lane = tid & 31;
  const int wave = tid >> 5;
  const int lh = lane & 15;
  const int hi = lane >> 4;
  const int ko = hi * 2;

  const float log_n = 6.9314718056f;  // log(1024)
  const float lb = -log_n;

  const float* xg = particles + (size_t)b * NN * DDIM;
  const float* lwg = logw_g + (size_t)b * NN;

  // warm the L2 path for this batch's inputs (global_prefetch_b8)
  __builtin_prefetch(xg + tid * 16, 0, 1);
  __builtin_prefetch(lwg + tid * 4, 0, 1);

  // ---- Phase 0: load, stats -------------------------------------------------
  float psum[4] = {0, 0, 0, 0}, psq[4] = {0, 0, 0, 0};
  float laM = -FBIG;
  for (int r = tid; r < NN; r += NT) {
    const float4 x = *(const float4*)(xg + r * 4);
    sxl[r * 4 + 0] = x.x;
    sxl[r * 4 + 1] = x.y;
    sxl[r * 4 + 2] = x.z;
    sxl[r * 4 + 3] = x.w;
    psum[0] += x.x;  psq[0] += x.x * x.x;
    psum[1] += x.y;  psq[1] += x.y * x.y;
    psum[2] += x.z;  psq[2] += x.z * x.z;
    psum[3] += x.w;  psq[3] += x.w * x.w;
    const float lw = lwg[r];
    la[r] = lw;
    laM = fmaxf(laM, lw);
  }
  float mean_d[4];
  float dia = 0.f;
#pragma unroll
  for (int d = 0; d < 4; ++d) {
    const float s = block_sum(psum[d], scr, tid);
    const float sq = block_sum(psq[d], scr, tid);
    const float m = s * (1.0f / NN);
    float var = sq * (1.0f / NN) - m * m;
    var = fmaxf(var, 0.f);
    mean_d[d] = m;
    dia = fmaxf(dia, sqrtf(var));
  }
  laM = block_max(laM, scr, tid);
  if (dia == 0.f) dia = 1.f;
  const float scale = fmaxf(dia * 2.0f, 1e-8f);  // dia * sqrt(D), D=4
  const float invs = 1.0f / scale;

  // normalize in place, squared norms, global min/max, fbuf := lb
  float pmax = -FBIG, pmin = FBIG;
  for (int r = tid; r < NN; r += NT) {
    float n2 = 0.f;
#pragma unroll
    for (int d = 0; d < 4; ++d) {
      float v = (sxl[r * 4 + d] - mean_d[d]) * invs;
      sxl[r * 4 + d] = v;
      n2 += v * v;
      pmax = fmaxf(pmax, v);
      pmin = fminf(pmin, v);
    }
    nl[r] = n2;
    fbuf[r] = lb;
  }
  const float gmax = block_max(pmax, scr, tid);
  const float gmin = -block_max(-pmin, scr, tid);
  const float mm = gmax - gmin;
  const float eps0 = fmaxf(mm * mm, 1e-8f);

  // ---- Phase 1: initial potentials -----------------------------------------
  softmin_lds(sxl, nl, la, laM, ay, eps0, tid);
  __syncthreads();
  softmin_lds(sxl, nl, fbuf, lb, bx, eps0, tid);
  __syncthreads();
  for (int r = tid; r < NN; r += NT) {
    ax[r] = ay[r];
    by[r] = bx[r];
  }
  __syncthreads();

  // ---- Phase 2: symmetric Sinkhorn with eps-scaling -------------------------
  float re = eps0;
  const float sf = SCALING_F * SCALING_F;
  for (int it = 0; it < MAX_ITER_I - 1; ++it) {
    const float inv_re = 1.0f / re;

    float fM = -FBIG;
    for (int r = tid; r < NN; r += NT) {
      const float v = la[r] + bx[r] * inv_re;
      fbuf[r] = v;
      fM = fmaxf(fM, v);
    }
    fM = block_max(fM, scr, tid);
    softmin_lds(sxl, nl, fbuf, fM, t1, re, tid);  // aty(old bx)
    __syncthreads();

    fM = -FBIG;
    for (int r = tid; r < NN; r += NT) {
      const float v = lb + ay[r] * inv_re;
      fbuf[r] = v;
      fM = fmaxf(fM, v);
    }
    fM = block_max(fM, scr, tid);
    softmin_lds(sxl, nl, fbuf, fM, t2, re, tid);  // btx(old ay)
    __syncthreads();

    float ad = 0.f, bd = 0.f;
    for (int r = tid; r < NN; r += NT) {
      const float na = 0.5f * (ay[r] + t1[r]);
      ad = fmaxf(ad, fabsf(na - ay[r]));
      ay[r] = na;
      const float nb = 0.5f * (bx[r] + t2[r]);
      bd = fmaxf(bd, fabsf(nb - bx[r]));
      bx[r] = nb;
    }
    ad = block_max(ad, scr, tid);
    bd = block_max(bd, scr, tid);

    fM = -FBIG;
    for (int r = tid; r < NN; r += NT) {
      const float v = la[r] + ax[r] * inv_re;
      fbuf[r] = v;
      fM = fmaxf(fM, v);
    }
    fM = block_max(fM, scr, tid);
    softmin_lds(sxl, nl, fbuf, fM, t1, re, tid);  // atx(old ax)
    __syncthreads();

    fM = -FBIG;
    for (int r = tid; r < NN; r += NT) {
      const float v = lb + by[r] * inv_re;
      fbuf[r] = v;
      fM = fmaxf(fM, v);
    }
    fM = block_max(fM, scr, tid);
    softmin_lds(sxl, nl, fbuf, fM, t2, re, tid);  // bty(old by)
    __syncthreads();

    for (int r = tid; r < NN; r += NT) {
      ax[r] = 0.5f * (ax[r] + t1[r]);
      by[r] = 0.5f * (by[r] + t2[r]);
    }
    __syncthreads();

    const float ne = fmaxf(re * sf, EPS_F);
    const bool cont = (ne < re) | (ad > THRESH_F) | (bd > THRESH_F);
    re = ne;
    if (!cont) break;  // block-uniform (ad/bd broadcast)
  }

  // ---- Phase 3: alpha / beta at final eps -----------------------------------
  const float feps = EPS_F;
  const float inv_eps = 1.0f / feps;
  float fM = -FBIG;
  for (int r = tid; r < NN; r += NT) {
    const float v = la[r] + bx[r] * inv_eps;
    fbuf[r] = v;
    fM = fmaxf(fM, v);
  }
  fM = block_max(fM, scr, tid);
  softmin_lds(sxl, nl, fbuf, fM, t1, feps, tid);  // alpha
  __syncthreads();
  fM = -FBIG;
  for (int r = tid; r < NN; r += NT) {
    const float v = lb + ay[r] * inv_eps;
    fbuf[r] = v;
    fM = fmaxf(fM, v);
  }
  fM = block_max(fM, scr, tid);
  softmin_lds(sxl, nl, fbuf, fM, t2, feps, tid);  // beta
  __syncthreads();

  float am = -FBIG;
  for (int r = tid; r < NN; r += NT) am = fmaxf(am, t1[r]);
  const float amax = block_max(am, scr, tid);

  // ---- Phase 4: column log-normalizer: fbuf[j] = lse_i((a_i+b_j-C_ij)/eps) --
  for (int jb = wave; jb < NIB; jb += NWAVE) {
    const int col = jb * 16 + lh;
    v2f bf;
    bf.x = sxl[col * 4 + ko];
    bf.y = sxl[col * 4 + ko + 1];
    const float ncol = nl[col];
    const float betac = t2[col];
    const float Mj = (amax + betac) * inv_eps;  // cost >= 0 -> valid stabilizer
    float ssum = 0.f;
    for (int ib = 0; ib < NIB; ++ib) {
      const int arow = ib * 16 + lh;
      v2f a;
      a.x = sxl[arow * 4 + ko];
      a.y = sxl[arow * 4 + ko + 1];
      v8f g = {0.f, 0.f, 0.f, 0.f, 0.f, 0.f, 0.f, 0.f};
      g = wmma_gram(a, bf, g);
      const int rbase = ib * 16 + hi * 8;
#pragma unroll
      for (int v = 0; v < 8; ++v) {
        const float cost = 0.5f * (nl[rbase + v] + ncol) - g[v];
        const float tv = (t1[rbase + v] + betac - cost) * inv_eps;
        ssum += __expf(tv - Mj);
      }
    }
    ssum += __shfl_xor(ssum, 16, 32);  // merge the two row-halves
    if (hi == 0) fbuf[col] = Mj + __logf(ssum);
  }
  __syncthreads();

  // ---- Phase 5: transported = T @ x, T column-normalized --------------------
  const unsigned char flg = flags[b];
  for (int ib = wave; ib < NIB; ib += NWAVE) {
    const int i0 = ib * 16;
    const int arow = i0 + lh;
    v2f a;
    a.x = sxl[arow * 4 + ko];
    a.y = sxl[arow * 4 + ko + 1];
    const int rbase = i0 + hi * 8;
    float alr[8], nrow[8];
#pragma unroll
    for (int v = 0; v < 8; ++v) {
      alr[v] = t1[rbase + v];
      nrow[v] = nl[rbase + v];
    }
    float acc[8][4];
#pragma unroll
    for (int v = 0; v < 8; ++v)
#pragma unroll
      for (int d = 0; d < 4; ++d) acc[v][d] = 0.f;

    for (int jb = 0; jb < NIB; ++jb) {
      const int col = jb * 16 + lh;
      v2f bf;
      bf.x = sxl[col * 4 + ko];
      bf.y = sxl[col * 4 + ko + 1];
      v8f g = {0.f, 0.f, 0.f, 0.f, 0.f, 0.f, 0.f, 0.f};
      g = wmma_gram(a, bf, g);
      const float ncol = nl[col];
      const float addc = t2[col] * inv_eps - fbuf[col] + log_n + la[col];
      const float4 xc = *(const float4*)(xg + col * 4);  // original particles
#pragma unroll
      for (int v = 0; v < 8; ++v) {
        const float cost = 0.5f * (nrow[v] + ncol) - g[v];
        const float T = __expf((alr[v] - cost) * inv_eps + addc);
        acc[v][0] += T * xc.x;
        acc[v][1] += T * xc.y;
        acc[v][2] += T * xc.z;
        acc[v][3] += T * xc.w;
      }
    }
#pragma unroll
    for (int off = 8; off; off >>= 1) {
#pragma unroll
      for (int v = 0; v < 8; ++v)
#pragma unroll
        for (int d = 0; d < 4; ++d)
          acc[v][d] += __shfl_xor(acc[v][d], off, 32);
    }
    if (lh == 0) {
#pragma unroll
      for (int v = 0; v < 8; ++v) {
        const int row = rbase + v;
        float* op = out + ((size_t)b * NN + row) * 4;
        const float4 xo = *(const float4*)(xg + row * 4);
        op[0] = flg ? acc[v][0] : xo.x;
        op[1] = flg ? acc[v][1] : xo.y;
        op[2] = flg ? acc[v][2] : xo.z;
        op[3] = flg ? acc[v][3] : xo.w;
      }
    }
  }

  // ---- Phase 6: rw / rlw ----------------------------------------------------
  float* rw = out + (size_t)BB * NN * DDIM;
  float* rlw = rw + (size_t)BB * NN;
  const float* wg = weights + (size_t)b * NN;
  for (int r = tid; r < NN; r += NT) {
    rw[(size_t)b * NN + r] = flg ? (1.0f / NN) : wg[r];
    rlw[(size_t)b * NN + r] = flg ? (-log_n) : la[r];
  }
}

extern "C" void kernel_launch(void* const* d_in, const int* in_sizes, int n_in,
                              void* d_out, int out_size, void* d_ws,
                              size_t ws_size, hipStream_t stream) {
  (void)in_sizes; (void)n_in; (void)d_ws; (void)ws_size; (void)out_size;
  const float* particles = (const float*)d_in[0];
  const float* weights = (const float*)d_in[1];
  const float* logw = (const float*)d_in[2];
  const unsigned char* flags = (const unsigned char*)d_in[3];
  float* out = (float*)d_out;
  ot_sinkhorn_kernel<<<dim3(BB), dim3(NT), 0, stream>>>(particles, weights,
                                                        logw, flags, out);
}